// KeepHighestSemanticPointcloud_17738214932556
// MI455X (gfx1250) — compile-verified
//
#include <hip/hip_runtime.h>

// ---- compile-time constants mirroring the reference ----
#define RES_HALF   0.025f          // RES / 2
#define B_CONST    4
#define GRID_CONST 802             // int(20.0 / 0.025) + 2
#define NUM_CELLS  ((B_CONST + 1) * GRID_CONST * GRID_CONST + GRID_CONST)  // 3,216,822

// Quantize exactly like jnp.round(v / (RES/2)): IEEE f32 divide + round-half-even
// (v_rndne_f32 under the default RNE mode).
__device__ __forceinline__ int quant(float v) {
    return (int)__builtin_rintf(v / RES_HALF);
}

// Order-preserving unsigned encoding of an IEEE f32 (u32 compare == float compare).
__device__ __forceinline__ unsigned int fkey(float f) {
    unsigned int b = __float_as_uint(f);
    return (b & 0x80000000u) ? ~b : (b | 0x80000000u);
}

// Packed 64-bit key: float max, tie-break to smallest point index.
__device__ __forceinline__ unsigned long long pack_key(float h, int idx) {
    return ((unsigned long long)fkey(h) << 32) |
           (unsigned long long)(unsigned int)(0x7FFFFFFF - idx);
}

// ---------------- Pass 0: zero cell keys (128-bit stores) + init scalars ------
__global__ void __launch_bounds__(256)
k_init(uint4* __restrict__ cellKey16, int* __restrict__ scal) {
    // NUM_CELLS * 8 bytes == 1,608,411 uint4 elements (divides exactly).
    const int n16 = NUM_CELLS / 2;  // NUM_CELLS u64 == NUM_CELLS/2 uint4
    int i = blockIdx.x * blockDim.x + threadIdx.x;
    int stride = gridDim.x * blockDim.x;
    uint4 z = {0u, 0u, 0u, 0u};
    for (int c = i; c < n16; c += stride) cellKey16[c] = z;
    if (i == 0) {
        scal[0] = 0x7FFFFFFF;       // qz min
        scal[1] = (int)0x80000000;  // qz max
        scal[2] = 0x7FFFFFFF;       // qx min
        scal[3] = (int)0x80000000;  // qx max
    }
}

// ---------------- Pass 1: global min/max of quantized z (rows) and x (cols) ---
__global__ void __launch_bounds__(256)
k_minmax(const float* __restrict__ xyz, int n, int* __restrict__ scal) {
    int tid = blockIdx.x * blockDim.x + threadIdx.x;
    int stride = gridDim.x * blockDim.x;

    int zmn = 0x7FFFFFFF, zmx = (int)0x80000000;
    int xmn = 0x7FFFFFFF, xmx = (int)0x80000000;

    for (int i = tid; i < n; i += stride) {
        // speculative prefetch of next strided element (dropped if OOB)
        __builtin_prefetch(&xyz[3 * (i + stride)], 0, 0);
        float x = xyz[3 * i + 0];
        float z = xyz[3 * i + 2];
        int qx = quant(x);
        int qz = quant(z);
        zmn = min(zmn, qz); zmx = max(zmx, qz);
        xmn = min(xmn, qx); xmx = max(xmx, qx);
    }

    // wave32 butterfly reduction
    #pragma unroll
    for (int off = 16; off > 0; off >>= 1) {
        zmn = min(zmn, __shfl_xor(zmn, off, 32));
        zmx = max(zmx, __shfl_xor(zmx, off, 32));
        xmn = min(xmn, __shfl_xor(xmn, off, 32));
        xmx = max(xmx, __shfl_xor(xmx, off, 32));
    }

    // cross-wave reduction via LDS (256 threads = 8 wave32 on CDNA5)
    __shared__ int s[4][8];
    int wave = threadIdx.x >> 5;
    int lane = threadIdx.x & 31;
    if (lane == 0) {
        s[0][wave] = zmn; s[1][wave] = zmx;
        s[2][wave] = xmn; s[3][wave] = xmx;
    }
    __syncthreads();
    if (threadIdx.x == 0) {
        int a = 0x7FFFFFFF, b = (int)0x80000000;
        int c = 0x7FFFFFFF, d = (int)0x80000000;
        #pragma unroll
        for (int w = 0; w < 8; ++w) {
            a = min(a, s[0][w]); b = max(b, s[1][w]);
            c = min(c, s[2][w]); d = max(d, s[3][w]);
        }
        atomicMin(&scal[0], a); atomicMax(&scal[1], b);
        atomicMin(&scal[2], c); atomicMax(&scal[3], d);
    }
}

// Shared flat-index computation (scalars are uniform -> s_load + SALU math).
__device__ __forceinline__ int flat_index(const float* __restrict__ xyz,
                                          const int* __restrict__ batch,
                                          const int* __restrict__ scal,
                                          int i, float& height) {
    int qzmin = scal[0], qzmax = scal[1];
    int qxmin = scal[2], qxmax = scal[3];
    int rmax = qzmax - qzmin;           // rows.max() after shift
    int cmax = qxmax - qxmin;           // cols.max() after shift
    float x = xyz[3 * i + 0];
    height  = xyz[3 * i + 1];
    float z = xyz[3 * i + 2];
    int row = quant(z) - qzmin;
    int col = quant(x) - qxmin;
    return batch[i] * (rmax * cmax) + row * cmax + col;
}

// ---------------- Pass 2: fused max + argmin via one 64-bit atomicMax ---------
__global__ void __launch_bounds__(256)
k_scatter(const float* __restrict__ xyz, const int* __restrict__ batch,
          const int* __restrict__ scal, unsigned long long* __restrict__ cellKey,
          int n) {
    int i = blockIdx.x * blockDim.x + threadIdx.x;
    if (i >= n) return;
    float h;
    int flat = flat_index(xyz, batch, scal, i, h);
    atomicMax(&cellKey[flat], pack_key(h, i));     // global_atomic_max_u64
}

// ---------------- Pass 3: emit kept_heights and keep mask ---------------------
__global__ void __launch_bounds__(256)
k_write(const float* __restrict__ xyz, const int* __restrict__ batch,
        const int* __restrict__ scal, const unsigned long long* __restrict__ cellKey,
        float* __restrict__ out_h, float* __restrict__ out_keep, int n) {
    int i = blockIdx.x * blockDim.x + threadIdx.x;
    if (i >= n) return;
    float h;
    int flat = flat_index(xyz, batch, scal, i, h);
    bool keep = (cellKey[flat] == pack_key(h, i)); // unique winner per cell
    out_h[i]    = keep ? h : 0.0f;
    out_keep[i] = keep ? 1.0f : 0.0f;
}

extern "C" void kernel_launch(void* const* d_in, const int* in_sizes, int n_in,
                              void* d_out, int out_size, void* d_ws, size_t ws_size,
                              hipStream_t stream) {
    const float* xyz   = (const float*)d_in[0];   // (N,3) f32
    const int*   batch = (const int*)d_in[1];     // (N,)  i32
    // d_in[2] = semantics: unused by the reference computation.
    int n = in_sizes[1];

    float* out_h    = (float*)d_out;              // kept_heights, N floats
    float* out_keep = (float*)d_out + n;          // keep mask,    N floats

    char* ws = (char*)d_ws;
    int* scal = (int*)ws;                                   // 4 ints (256B pad)
    unsigned long long* cellKey = (unsigned long long*)(ws + 256); // ~25.7 MB

    const int TPB = 256;
    int point_blocks = (n + TPB - 1) / TPB;

    k_init   <<<2048, TPB, 0, stream>>>((uint4*)cellKey, scal);
    k_minmax <<<2048, TPB, 0, stream>>>(xyz, n, scal);
    k_scatter<<<point_blocks, TPB, 0, stream>>>(xyz, batch, scal, cellKey, n);
    k_write  <<<point_blocks, TPB, 0, stream>>>(xyz, batch, scal, cellKey,
                                                out_h, out_keep, n);
}